// SocialPool_46385646796879
// MI455X (gfx1250) — compile-verified
//
#include <hip/hip_runtime.h>
#include <hip/hip_bf16.h>

// SocialPool for MI455X (gfx1250, wave32).
// Phase A (per block = 16 agents): bin all 4096 neighbors into 48 log-polar
//   segments using pure compare trees (no logf/atan2f), accumulate hidden
//   rows into 16x48x64 f32 LDS bins via ds_add_f32 (bank-conflict-free layout
//   [agent][channel*48+seg]), count pairs with ds_add_u32.
// Phase B: out(16x48) = (S * inv_cnt)(16x3072) @ fc_w^T(3072x48) using
//   V_WMMA_F32_16X16X4_F32 (exact f32 math), 8 waves split K, partials
//   reduced in LDS, then bias + relu.

#define NSEG      48
#define N_AGENTS  4096
#define HIDDEN    64
#define OUT_DIM   48
#define FAN_IN    (NSEG * HIDDEN)   // 3072
#define AG_PER_BLK 16
#define THREADS   256               // 8 wave32

typedef __attribute__((ext_vector_type(2))) float v2f;
typedef __attribute__((ext_vector_type(8))) float v8f;

// Squared ring thresholds t_k = 0.5 * 8^(k/5), k = 0..6 (computed in double,
// rounded): compare r^2 directly, no sqrt/log needed.
#define T1SQ 0.5743491775f
#define T2SQ 1.3195079108f
#define T3SQ 3.0314331330f
#define T4SQ 6.9644045064f
#define T5SQ 16.0f
#define T6SQ 36.75834736f

__global__ __launch_bounds__(THREADS)
void social_pool_kernel(const float* __restrict__ ypred,
                        const float* __restrict__ hidden,
                        const float* __restrict__ fc_w,
                        const float* __restrict__ fc_b,
                        float* __restrict__ out)
{
    // 201 KB of LDS: legal on CDNA5 (320 KB per workgroup).
    __shared__ float S[AG_PER_BLK * FAN_IN];        // [agent][c*48 + s]
    __shared__ int   cnt[AG_PER_BLK * NSEG];
    __shared__ float cinv[AG_PER_BLK * NSEG];
    __shared__ float outAcc[AG_PER_BLK * OUT_DIM];

    const int tid  = threadIdx.x;
    const int lane = tid & 31;
    const int wv   = tid >> 5;                      // wave id 0..7
    const int i0   = blockIdx.x * AG_PER_BLK;

    for (int k = tid; k < AG_PER_BLK * FAN_IN; k += THREADS) S[k] = 0.0f;
    for (int k = tid; k < AG_PER_BLK * NSEG;   k += THREADS) cnt[k] = 0;
    for (int k = tid; k < AG_PER_BLK * OUT_DIM; k += THREADS) outAcc[k] = 0.0f;
    __syncthreads();

    // Each lane owns agent (lane & 15); lanes 16..31 duplicate 0..15 so that
    // readlane(seg, a) is valid for a in [0,16).
    const int   al = lane & 15;
    const float xi = ypred[2 * (i0 + al) + 0];
    const float yi = ypred[2 * (i0 + al) + 1];

    const int jcnt = N_AGENTS / 8;
    const int jb   = wv * jcnt;
    for (int j = jb; j < jb + jcnt; ++j) {
        const float xj = ypred[2 * j + 0];
        const float yj = ypred[2 * j + 1];
        // theta direction in reference: yp[j] - yp[i]; r is symmetric.
        const float dx = xj - xi;
        const float dy = yj - yi;
        const float r2 = dx * dx + dy * dy;

        int ring = (r2 >= T1SQ) + (r2 >= T2SQ) + (r2 >= T3SQ)
                 + (r2 >= T4SQ) + (r2 >= T5SQ);

        // wedge = mod(trunc(theta*4/pi + 3), 8) via exact sign/diagonal tests
        int wedge;
        if (dy > 0.0f)      wedge = (dx > 0.0f) ? ((dy < dx) ? 3 : 4)
                                                : ((dy > -dx) ? 5 : 6);
        else if (dy < 0.0f) wedge = (dx < 0.0f) ? 0 : ((-dy > dx) ? 1 : 2);
        else                wedge = (dx > 0.0f) ? 3 : 7;  // dy==0 (dx==0 -> invalid anyway)

        const bool valid = (r2 >= 0.25f) && (r2 < T6SQ);
        const int  seg   = valid ? (ring * 8 + wedge) : -1;

        // one count per (agent, j) pair: lanes 0..15 each handle one agent
        if (lane < 16 && seg >= 0) atomicAdd(&cnt[al * NSEG + seg], 1);

        // hidden row j: lane covers channels lane and lane+32 (coalesced)
        const float h0 = hidden[j * HIDDEN + lane];
        const float h1 = hidden[j * HIDDEN + 32 + lane];
        __builtin_prefetch(&hidden[(j + 8) * HIDDEN + lane], 0, 1);

#pragma unroll
        for (int a = 0; a < AG_PER_BLK; ++a) {
            const int s = __builtin_amdgcn_readlane(seg, a);  // uniform
            if (s >= 0) {
                // layout [agent][channel*48 + seg]: 32 lanes -> 32 banks
                atomicAdd(&S[a * FAN_IN + lane * NSEG + s], h0);
                atomicAdd(&S[a * FAN_IN + (lane + 32) * NSEG + s], h1);
            }
        }
    }
    __syncthreads();

    // inv-count (folds mean division + where(cnt>0,...,0) into A fetch)
    for (int k = tid; k < AG_PER_BLK * NSEG; k += THREADS) {
        const int c = cnt[k];
        cinv[k] = (c > 0) ? (1.0f / (float)c) : 0.0f;
    }
    __syncthreads();

    // WMMA phase: out_pre(16x48) = A(16x3072) @ B(3072x48)
    // A[m,f] = S[m][ (f&63)*48 + (f>>6) ] * cinv[m][f>>6],  B[k,n] = fc_w[n*3072 + k]
    v8f acc0 = {0.f,0.f,0.f,0.f,0.f,0.f,0.f,0.f};
    v8f acc1 = acc0, acc2 = acc0;

    const int m    = lane & 15;            // A row / B col / D col
    const int koff = (lane >> 4) << 1;     // lanes 0-15: K0,K1; lanes 16-31: K2,K3
    const int kb0  = wv * (FAN_IN / 8);    // per-wave K slice (384 features)

    for (int kb = kb0; kb < kb0 + FAN_IN / 8; kb += 4) {
        const int kk = kb + koff;          // even, and kk/kk+1 share a segment
        const int s  = kk >> 6;
        const int c  = kk & 63;
        const float sc = cinv[m * NSEG + s];
        v2f a;
        a.x = S[m * FAN_IN + c * NSEG + s] * sc;
        a.y = S[m * FAN_IN + (c + 1) * NSEG + s] * sc;

        const v2f b0 = *(const v2f*)(fc_w + (size_t)(0 * 16 + m) * FAN_IN + kk);
        acc0 = __builtin_amdgcn_wmma_f32_16x16x4_f32(false, a, false, b0,
                                                     (short)0, acc0, false, false);
        const v2f b1 = *(const v2f*)(fc_w + (size_t)(1 * 16 + m) * FAN_IN + kk);
        acc1 = __builtin_amdgcn_wmma_f32_16x16x4_f32(false, a, false, b1,
                                                     (short)0, acc1, false, false);
        const v2f b2 = *(const v2f*)(fc_w + (size_t)(2 * 16 + m) * FAN_IN + kk);
        acc2 = __builtin_amdgcn_wmma_f32_16x16x4_f32(false, a, false, b2,
                                                     (short)0, acc2, false, false);
    }

    // C/D layout: VGPR v holds (M = v + 8*(lane>=16), N = lane&15)
    const int mrow = (lane >> 4) << 3;
#pragma unroll
    for (int v = 0; v < 8; ++v) {
        const int row = mrow + v;
        atomicAdd(&outAcc[row * OUT_DIM +  0 + m], acc0[v]);
        atomicAdd(&outAcc[row * OUT_DIM + 16 + m], acc1[v]);
        atomicAdd(&outAcc[row * OUT_DIM + 32 + m], acc2[v]);
    }
    __syncthreads();

    for (int k = tid; k < AG_PER_BLK * OUT_DIM; k += THREADS) {
        const int o = k % OUT_DIM;
        const float val = outAcc[k] + fc_b[o];
        out[(size_t)(i0 + k / OUT_DIM) * OUT_DIM + o] = val > 0.0f ? val : 0.0f;
    }
}

extern "C" void kernel_launch(void* const* d_in, const int* in_sizes, int n_in,
                              void* d_out, int out_size, void* d_ws, size_t ws_size,
                              hipStream_t stream) {
    const float* ypred  = (const float*)d_in[0];
    const float* hidden = (const float*)d_in[1];
    const float* fc_w   = (const float*)d_in[2];
    const float* fc_b   = (const float*)d_in[3];
    float* out = (float*)d_out;
    (void)in_sizes; (void)n_in; (void)out_size; (void)d_ws; (void)ws_size;

    social_pool_kernel<<<dim3(N_AGENTS / AG_PER_BLK), dim3(THREADS), 0, stream>>>(
        ypred, hidden, fc_w, fc_b, out);
}